// SAGEConv_27410481283335
// MI455X (gfx1250) — compile-verified
//
#include <hip/hip_runtime.h>

typedef __attribute__((ext_vector_type(2))) float v2f;
typedef __attribute__((ext_vector_type(8))) float v8f;

#define N_NODES 50000
#define N_EDGES 800000
#define D 64            // D_IN == D_OUT == 64
#define M_TILES (N_NODES / 16)   // 3125, exact

// ---------------------------------------------------------------------------
// Zero workspace regions (agg + deg are contiguous)
// ---------------------------------------------------------------------------
__global__ void sage_zero_kernel(float* __restrict__ p, int n) {
    int i = blockIdx.x * blockDim.x + threadIdx.x;
    if (i < n) p[i] = 0.0f;
}

// ---------------------------------------------------------------------------
// Fused dual GEMM via V_WMMA_F32_16X16X4_F32:
//   srcdata = feat @ W_neigh^T          (to workspace)
//   out     = feat @ W_self^T + b_self  (to d_out)
// One wave computes a 16(M) x 64(N) slab; K=64 in 16 steps of 4.
// A-frag: lane L -> row M = L%16, K pair at kh=(L/16)*2.
// B-frag (col-major KxN): lane L -> col N = L%16, same K pair.
//   B[k][n] = W[n][k]  (we need W^T as the B operand).
// C/D:   lane L, vgpr v -> M = (L/16)*8 + v, N = L%16.
// ---------------------------------------------------------------------------
__global__ __launch_bounds__(256) void sage_gemm_kernel(
    const float* __restrict__ feat,
    const float* __restrict__ Wn,
    const float* __restrict__ Ws,
    const float* __restrict__ bias,
    float* __restrict__ srcdata,
    float* __restrict__ out)
{
    const int wave = threadIdx.x >> 5;            // 8 waves / block
    const int lane = threadIdx.x & 31;
    const int tile = blockIdx.x * 8 + wave;       // M-tile index
    if (tile >= M_TILES) return;                  // uniform across the wave

    const int m  = tile * 16 + (lane & 15);       // A row
    const int kh = (lane >> 4) * 2;               // K sub-offset (0 or 2)
    const int n0 = lane & 15;                     // column within an N-tile

    v8f accN[4] = {};
    v8f accS[4] = {};

#pragma unroll
    for (int kk = 0; kk < 16; ++kk) {
        const int kb = kk * 4 + kh;
        v2f a;
        a.x = feat[m * D + kb];
        a.y = feat[m * D + kb + 1];
#pragma unroll
        for (int t = 0; t < 4; ++t) {
            const int n = t * 16 + n0;
            v2f bn, bs;
            bn.x = Wn[n * D + kb];
            bn.y = Wn[n * D + kb + 1];
            bs.x = Ws[n * D + kb];
            bs.y = Ws[n * D + kb + 1];
            accN[t] = __builtin_amdgcn_wmma_f32_16x16x4_f32(
                false, a, false, bn, (short)0, accN[t], false, false);
            accS[t] = __builtin_amdgcn_wmma_f32_16x16x4_f32(
                false, a, false, bs, (short)0, accS[t], false, false);
        }
    }

    const int rowBase = tile * 16 + (lane >> 4) * 8;
#pragma unroll
    for (int t = 0; t < 4; ++t) {
        const int col = t * 16 + n0;
        const float b = bias[col];
#pragma unroll
        for (int v = 0; v < 8; ++v) {
            const int r = rowBase + v;
            srcdata[r * D + col] = accN[t][v];
            out[r * D + col]     = accS[t][v] + b;
        }
    }
}

// ---------------------------------------------------------------------------
// Edge scatter: agg[dst] += srcdata[src]; deg[dst] += 1
// 16 threads per edge, float4 per thread, HW fp32 global atomics.
// ---------------------------------------------------------------------------
__global__ __launch_bounds__(256) void sage_edge_kernel(
    const int* __restrict__ src,
    const int* __restrict__ dst,
    const float* __restrict__ srcdata,
    float* __restrict__ agg,
    float* __restrict__ deg)
{
    const int e = blockIdx.x * 16 + (threadIdx.x >> 4);   // edge index
    if (e >= N_EDGES) return;
    const int dbase = (threadIdx.x & 15) * 4;             // dim base (0..60)
    const int s = src[e];
    const int t = dst[e];

    const float4 val = *(const float4*)(srcdata + (size_t)s * D + dbase);
    float* ap = agg + (size_t)t * D + dbase;
    unsafeAtomicAdd(ap + 0, val.x);
    unsafeAtomicAdd(ap + 1, val.y);
    unsafeAtomicAdd(ap + 2, val.z);
    unsafeAtomicAdd(ap + 3, val.w);
    if (dbase == 0) unsafeAtomicAdd(deg + t, 1.0f);
}

// ---------------------------------------------------------------------------
// Finalize: out += (deg > 0) ? agg / deg : 0
// ---------------------------------------------------------------------------
__global__ void sage_finalize_kernel(
    const float* __restrict__ agg,
    const float* __restrict__ deg,
    float* __restrict__ out)
{
    int i = blockIdx.x * blockDim.x + threadIdx.x;
    if (i >= N_NODES * D) return;
    const float dg = deg[i >> 6];
    if (dg > 0.0f) out[i] += agg[i] / dg;
}

// ---------------------------------------------------------------------------
extern "C" void kernel_launch(void* const* d_in, const int* in_sizes, int n_in,
                              void* d_out, int out_size, void* d_ws, size_t ws_size,
                              hipStream_t stream)
{
    const float* feat = (const float*)d_in[0];
    const int*   src  = (const int*)d_in[1];
    const int*   dst  = (const int*)d_in[2];
    const float* Wn   = (const float*)d_in[3];
    const float* Ws   = (const float*)d_in[4];
    const float* bias = (const float*)d_in[5];
    float* out = (float*)d_out;

    // workspace layout: srcdata | agg | deg   (~25.8 MB)
    float* srcdata = (float*)d_ws;
    float* agg     = srcdata + (size_t)N_NODES * D;
    float* deg     = agg     + (size_t)N_NODES * D;

    // 1) zero agg + deg (contiguous)
    const int zn = N_NODES * D + N_NODES;
    sage_zero_kernel<<<(zn + 255) / 256, 256, 0, stream>>>(agg, zn);

    // 2) fused dual WMMA GEMM
    sage_gemm_kernel<<<(M_TILES + 7) / 8, 256, 0, stream>>>(
        feat, Wn, Ws, bias, srcdata, out);

    // 3) edge scatter (16 edges per 256-thread block)
    sage_edge_kernel<<<(N_EDGES + 15) / 16, 256, 0, stream>>>(
        src, dst, srcdata, agg, deg);

    // 4) finalize mean + add
    sage_finalize_kernel<<<(N_NODES * D + 255) / 256, 256, 0, stream>>>(
        agg, deg, out);
}